// CRF_86706799772067
// MI455X (gfx1250) — compile-verified
//
#include <hip/hip_runtime.h>
#include <stdint.h>

typedef __attribute__((ext_vector_type(2))) float v2f;
typedef __attribute__((ext_vector_type(8))) float v8f;

#define HSZ 512
#define WSZ 512
#define HW  (512*512)
#define CCH 8
#define BB  4
#define BC  32
#define TILE 16

// Bilateral: reference radius is 11, but with SPATIAL_SIGMA=1 all taps beyond
// radius 6 have weight <= exp(-24.5) ~ 2.3e-11 against a denominator >= 1
// (center tap weight == 1) -> below f32 resolution. Use a 13x13 window.
#define BRE 6
#define BWE 13
#define NTAPSE (BWE*BWE)        // 169
#define BTILEE (TILE + 2*BRE)   // 28

#define QROWS 528               // 5 top + 512 + 11 bottom zero rows (28-row WMMA window)
#define TCOLS 528               // 8 left + 512 + 8 right zero cols

// ---------------- zero the pad borders of qpad / tpad (once per launch) ----------------
__global__ __launch_bounds__(256) void zero_pads(float* __restrict__ qpad,
                                                 float* __restrict__ tpad) {
  int idx = blockIdx.x * 256 + threadIdx.x;       // 262144 threads
  {
    int bc  = idx >> 13;                          // 16 rows * 512 cols per bc
    int r   = (idx >> 9) & 15;
    int col = idx & 511;
    int row = (r < 5) ? r : (512 + r);            // rows 0..4 and 517..527
    qpad[((size_t)bc * QROWS + row) * 512 + col] = 0.f;
  }
  {
    int bc  = idx >> 13;                          // 512 rows * 16 cols per bc
    int r   = (idx >> 4) & 511;
    int c   = idx & 15;
    int col = (c < 8) ? c : (512 + c);            // cols 0..7 and 520..527
    tpad[((size_t)bc * 512 + r) * TCOLS + col] = 0.f;
  }
}

// ---------------- bilateral filter (loop-invariant potential) ----------------
__global__ __launch_bounds__(256) void bilateral_kernel(const float* __restrict__ img,
                                                        float* __restrict__ fp) {
  __shared__ float s_tile[BTILEE * BTILEE];
  __shared__ float s_sw[NTAPSE];
  int block = blockIdx.x;                // 8 ch * 32 * 32 tiles
  int c  = block >> 10;
  int ty = (block >> 5) & 31;
  int tx = block & 31;
  int tid = threadIdx.x;
  const float* ic = img + c * HW;
  int y0 = ty * TILE - BRE, x0 = tx * TILE - BRE;
  // Async global->LDS DMA tile fill (ASYNCcnt path), edge-clamped addresses.
  for (int i = tid; i < BTILEE * BTILEE; i += 256) {
    int yy = y0 + i / BTILEE;
    int xx = x0 + i % BTILEE;
    yy = min(max(yy, 0), HSZ - 1);
    xx = min(max(xx, 0), WSZ - 1);
    const float* gsrc = ic + yy * WSZ + xx;
    uint32_t ldsoff = (uint32_t)(uintptr_t)(&s_tile[i]);
    asm volatile("global_load_async_to_lds_b32 %0, %1, off"
                 :: "v"(ldsoff), "v"(gsrc) : "memory");
  }
  for (int t = tid; t < NTAPSE; t += 256) {
    float dy = (float)(t / BWE - BRE), dx = (float)(t % BWE - BRE);
    s_sw[t] = __expf(-0.5f * (dy * dy + dx * dx));   // SPATIAL_SIGMA = 1
  }
  asm volatile("s_wait_asynccnt 0" ::: "memory");
  __syncthreads();
  int ly = tid >> 4, lx = tid & 15;
  float center = s_tile[(ly + BRE) * BTILEE + lx + BRE];
  float num = 0.f, den = 0.f;
#pragma unroll 1
  for (int dy = 0; dy < BWE; ++dy) {
    const float* rowp = &s_tile[(ly + dy) * BTILEE + lx];
    const float* swp  = &s_sw[dy * BWE];
#pragma unroll
    for (int dx = 0; dx < BWE; ++dx) {
      float sh = rowp[dx];
      float d = center - sh;
      float w = swp[dx] * __expf(-0.5f * d * d);     // RANGE_SIGMA = 1
      num += w * sh;
      den += w;
    }
  }
  fp[c * HW + (ty * TILE + ly) * WSZ + tx * TILE + lx] = num / den;
}

// ---------------- channel softmax -> row-padded q ----------------
__global__ __launch_bounds__(256) void softmax_kernel(const float* __restrict__ x,
                                                      float* __restrict__ qpad) {
  int idx = blockIdx.x * 256 + threadIdx.x;   // over B*H*W
  int b   = idx >> 18;
  int hw  = idx & (HW - 1);
  int row = hw >> 9, col = hw & 511;
  const float* xp = x + (size_t)b * CCH * HW + hw;
  float v[CCH];
  float m = -3.4e38f;
#pragma unroll
  for (int c = 0; c < CCH; ++c) { v[c] = xp[c * HW]; m = fmaxf(m, v[c]); }
  float s = 0.f;
#pragma unroll
  for (int c = 0; c < CCH; ++c) { v[c] = __expf(v[c] - m); s += v[c]; }
  float inv = 1.0f / s;
  float* qp = qpad + ((size_t)(b * CCH) * QROWS + row + 5) * 512 + col;
#pragma unroll
  for (int c = 0; c < CCH; ++c) qp[(size_t)c * QROWS * 512] = v[c] * inv;
}

// Gaussian tap value for t = jj - i; valid taps t in [0,10], center (t==5) zeroed.
__device__ __forceinline__ float gcoef(int t, float invth) {
  if (t < 0 || t > 10 || t == 5) return 0.f;
  float d = (float)(t - 5) * invth;
  return __expf(-0.5f * d * d);
}

// ---------------- H-direction 11-tap conv as V_WMMA_F32_16X16X4_F32 ----------------
// O(16x16) = K(16x28 band) x Z(28x16 rows of padded q). Each wave: 4 tiles along W,
// reusing the coefficient (A) operands across tiles. No guards: pads are zero.
__global__ __launch_bounds__(256) void conv_h_kernel(const float* __restrict__ qpad,
                                                     const float* __restrict__ invtheta,
                                                     float* __restrict__ tpad) {
  int wave = (blockIdx.x << 3) + (threadIdx.x >> 5);   // 8192 waves
  int lane = threadIdx.x & 31;
  int tg = wave & 7, th = (wave >> 3) & 31, bc = wave >> 8;
  float invth = invtheta[0];
  int half = lane >> 4;
  int mn = lane & 15;                     // A: output row m ; B: column n
  v2f a[7];
#pragma unroll
  for (int j = 0; j < 7; ++j) {
    int kb = 4 * j + 2 * half;
    a[j].x = gcoef(kb - mn, invth);       // A[m][kb] = kh[kb-m]
    a[j].y = gcoef(kb + 1 - mn, invth);
  }
  const float* qbase = qpad + ((size_t)bc * QROWS + th * TILE) * 512 + tg * 64 + mn;
  float* obase = tpad + ((size_t)bc * 512 + th * TILE + 8 * half) * TCOLS + tg * 64 + mn + 8;
#pragma unroll
  for (int t = 0; t < 4; ++t) {
    const float* qc = qbase + t * TILE;
    v8f acc = {};
#pragma unroll
    for (int j = 0; j < 7; ++j) {
      int kb = 4 * j + 2 * half;
      v2f b;
      b.x = qc[kb * 512];                 // Z[kb][n], padded rows are zero
      b.y = qc[kb * 512 + 512];
      acc = __builtin_amdgcn_wmma_f32_16x16x4_f32(false, a[j], false, b,
                                                  (short)0, acc, false, false);
    }
    float* op = obase + t * TILE;
#pragma unroll
    for (int v = 0; v < 8; ++v) op[v * TCOLS] = acc[v];
  }
}

// ---------------- W-direction conv + CRF update, fused ----------------
// O(16x16) = Z'(16x28 cols of padded tmp) x K'(28x16 band); x_new = x0 + w*O + fp.
// Each wave: 4 tiles along H, reusing the coefficient (B) operands across tiles.
__global__ __launch_bounds__(256) void conv_w_update_kernel(const float* __restrict__ tpad,
                                                            const float* __restrict__ x0,
                                                            const float* __restrict__ fp,
                                                            const float* __restrict__ invtheta,
                                                            const float* __restrict__ sweight,
                                                            float* __restrict__ out) {
  int wave = (blockIdx.x << 3) + (threadIdx.x >> 5);   // 8192 waves
  int lane = threadIdx.x & 31;
  int tw = wave & 31, thg = (wave >> 5) & 7, bc = wave >> 8;
  float invth = invtheta[1];
  float wsm = sweight[0];
  int half = lane >> 4;
  int mn = lane & 15;                     // A: row m ; B: column n
  v2f b[7];
#pragma unroll
  for (int j = 0; j < 7; ++j) {
    int kb = 4 * j + 2 * half;
    b[j].x = gcoef(kb - mn, invth);       // B[kb][n] = kw[kb-n]
    b[j].y = gcoef(kb + 1 - mn, invth);
  }
  int ch = bc & 7;
#pragma unroll
  for (int t = 0; t < 4; ++t) {
    int throw_ = (thg * 4 + t) * TILE;
    // col index for window position kb: (tw*16 - 5 + kb) + 8 = tw*16 + kb + 3
    const float* tc = tpad + ((size_t)bc * 512 + throw_ + mn) * TCOLS + tw * TILE + 3;
    v8f acc = {};
#pragma unroll
    for (int j = 0; j < 7; ++j) {
      int kb = 4 * j + 2 * half;
      v2f a;
      a.x = tc[kb];                       // adjacent cols -> b64 load
      a.y = tc[kb + 1];
      acc = __builtin_amdgcn_wmma_f32_16x16x4_f32(false, a, false, b[j],
                                                  (short)0, acc, false, false);
    }
    int base = (throw_ + 8 * half) * WSZ + tw * TILE + mn;
    const float* x0p = x0 + (size_t)bc * HW + base;
    const float* fpp = fp + (size_t)ch * HW + base;
    float* op = out + (size_t)bc * HW + base;
#pragma unroll
    for (int v = 0; v < 8; ++v)
      op[v * WSZ] = x0p[v * WSZ] + wsm * acc[v] + fpp[v * WSZ];
  }
}

extern "C" void kernel_launch(void* const* d_in, const int* in_sizes, int n_in,
                              void* d_out, int out_size, void* d_ws, size_t ws_size,
                              hipStream_t stream) {
  const float* x0       = (const float*)d_in[0];  // (4,8,512,512)
  const float* img      = (const float*)d_in[1];  // (8,512,512)
  const float* sweight  = (const float*)d_in[2];  // scalar
  const float* invtheta = (const float*)d_in[3];  // (2,)
  float* out = (float*)d_out;

  char* ws = (char*)d_ws;
  float* fp   = (float*)ws;                                          //  8.0 MB
  float* qpad = (float*)(ws + (size_t)CCH * HW * 4);                 // 33.0 MB (32 x 528 x 512)
  float* tpad = (float*)(ws + (size_t)CCH * HW * 4
                            + (size_t)BC * QROWS * 512 * 4);         // 33.0 MB (32 x 512 x 528)

  zero_pads<<<1024, 256, 0, stream>>>(qpad, tpad);
  bilateral_kernel<<<CCH * 32 * 32, 256, 0, stream>>>(img, fp);

  const float* xcur = x0;
  for (int it = 0; it < 5; ++it) {
    softmax_kernel<<<(BB * HW) / 256, 256, 0, stream>>>(xcur, qpad);
    conv_h_kernel<<<1024, 256, 0, stream>>>(qpad, invtheta, tpad);
    conv_w_update_kernel<<<1024, 256, 0, stream>>>(tpad, x0, fp,
                                                   invtheta, sweight, out);
    xcur = out;
  }
}